// SAGPool_30880814858412
// MI455X (gfx1250) — compile-verified
//
#include <hip/hip_runtime.h>
#include <math.h>

#define G 512
#define N_PER_G 256
#define K_TOP 64
#define NUM_NODES (G * N_PER_G)   // 131072
#define NUM_EDGES 524288
#define D 128
#define GK (G * K_TOP)            // 32768
#define EPSF 1e-6f

typedef float v2f __attribute__((ext_vector_type(2)));
typedef float v8f __attribute__((ext_vector_type(8)));

// ---------------- init: zero accumulators, rank = -1 ----------------
__global__ void init_kernel(float* __restrict__ deg, float* __restrict__ score,
                            int* __restrict__ rank, float* __restrict__ agg) {
    int i = blockIdx.x * blockDim.x + threadIdx.x;
    if (i < GK * D) agg[i] = 0.f;
    if (i < NUM_NODES) { deg[i] = 0.f; score[i] = 0.f; rank[i] = -1; }
}

// ---------------- weighted degree: deg[dst] += w ----------------
__global__ void deg_kernel(const int* __restrict__ ei, const float* __restrict__ w,
                           float* __restrict__ deg) {
    int e = blockIdx.x * blockDim.x + threadIdx.x;
    if (e >= NUM_EDGES) return;
    atomicAdd(&deg[ei[NUM_EDGES + e]], w[e]);
}

// ---------------- s[n] = x[n] . w_score  (wave per node) ----------------
__global__ void score_pre_kernel(const float* __restrict__ x,
                                 const float* __restrict__ wsc,
                                 float* __restrict__ s) {
    int lane = threadIdx.x & 31;
    int node = blockIdx.x * 8 + (threadIdx.x >> 5);     // 8 waves / block
    const float4 xv = *(const float4*)(x + (size_t)node * D + lane * 4);
    const float4 wv = *(const float4*)(wsc + lane * 4);
    float v = xv.x * wv.x + xv.y * wv.y + xv.z * wv.z + xv.w * wv.w;
    #pragma unroll
    for (int off = 16; off > 0; off >>= 1) v += __shfl_down(v, off, 32);
    if (lane == 0) s[node] = v;
}

// ---------------- score[dst] += norm * s[src] (scalar message pass) ----------------
__global__ void edge_score_kernel(const int* __restrict__ ei, const float* __restrict__ w,
                                  const float* __restrict__ deg, const float* __restrict__ s,
                                  float* __restrict__ score) {
    int e = blockIdx.x * blockDim.x + threadIdx.x;
    if (e >= NUM_EDGES) return;
    int src = ei[e], dst = ei[NUM_EDGES + e];
    float ds = fmaxf(deg[src], EPSF), dd = fmaxf(deg[dst], EPSF);
    float norm = w[e] * rsqrtf(ds * dd);
    atomicAdd(&score[dst], norm * s[src]);
}

// ---------------- per-graph bitonic top-K (desc score, tie: lower idx) ----------------
__global__ void topk_kernel(const float* __restrict__ score, int* __restrict__ rank,
                            float* __restrict__ gate) {
    __shared__ float sc[N_PER_G];
    __shared__ int   id[N_PER_G];
    int tid = threadIdx.x, g = blockIdx.x;
    sc[tid] = score[g * N_PER_G + tid];
    id[tid] = tid;
    __syncthreads();
    for (int k = 2; k <= N_PER_G; k <<= 1) {
        for (int j = k >> 1; j > 0; j >>= 1) {
            int i = tid, ixj = i ^ j;
            if (ixj > i) {
                bool asc = ((i & k) == 0);
                float si = sc[i], sj = sc[ixj];
                int   ii = id[i], ij = id[ixj];
                // "first" = higher score, or equal score with lower index (matches top_k)
                bool iFirst = (si > sj) || (si == sj && ii < ij);
                if (asc ? !iFirst : iFirst) {
                    sc[i] = sj; sc[ixj] = si;
                    id[i] = ij; id[ixj] = ii;
                }
            }
            __syncthreads();
        }
    }
    if (tid < K_TOP) {
        int node = g * N_PER_G + id[tid];
        int r    = g * K_TOP + tid;
        rank[node] = r;
        gate[r]    = tanhf(sc[tid]);
    }
}

// ---------------- selected-edge feature scatter (wave per edge) ----------------
__global__ void feat_scatter_kernel(const int* __restrict__ ei, const float* __restrict__ w,
                                    const float* __restrict__ deg, const float* __restrict__ x,
                                    const int* __restrict__ rank, float* __restrict__ agg) {
    int lane = threadIdx.x & 31;
    int e = blockIdx.x * 8 + (threadIdx.x >> 5);        // 8 waves / block
    int dst = ei[NUM_EDGES + e];
    int r = rank[dst];
    if (r < 0) return;                                  // uniform per wave
    int src = ei[e];
    float ds = fmaxf(deg[src], EPSF), dd = fmaxf(deg[dst], EPSF);
    float norm = w[e] * rsqrtf(ds * dd);
    const float4 xv = *(const float4*)(x + (size_t)src * D + lane * 4);
    float* o = agg + (size_t)r * D + lane * 4;
    atomicAdd(o + 0, norm * xv.x);
    atomicAdd(o + 1, norm * xv.y);
    atomicAdd(o + 2, norm * xv.z);
    atomicAdd(o + 3, norm * xv.w);
}

// ---------------- out = (agg @ W_feat) * gate, via V_WMMA_F32_16X16X4_F32 ----------------
// One wave computes a 16x16 tile. Block = 8 waves covering all 8 column tiles (N=128).
__global__ void gemm_gate_kernel(const float* __restrict__ agg, const float* __restrict__ Wf,
                                 const float* __restrict__ gate, float* __restrict__ out) {
    int lane = threadIdx.x & 31;
    int wave = threadIdx.x >> 5;
    int m0 = blockIdx.x * 16;
    int n0 = wave * 16;
    int col = lane & 15;
    int khalf = (lane >> 4) << 1;                       // 0 (lanes 0-15) or 2 (lanes 16-31)

    v8f c = {};
    const float* arow = agg + (size_t)(m0 + col) * D;   // A: M = lane&15, K pair per half-wave
    #pragma unroll 4
    for (int k0 = 0; k0 < D; k0 += 4) {
        int kb = k0 + khalf;
        v2f a, b;
        a.x = arow[kb];
        a.y = arow[kb + 1];
        b.x = Wf[(size_t)kb * D + n0 + col];            // B: row kb striped across lanes
        b.y = Wf[(size_t)(kb + 1) * D + n0 + col];
        c = __builtin_amdgcn_wmma_f32_16x16x4_f32(
                /*neg_a=*/false, a, /*neg_b=*/false, b,
                /*c_mod=*/(short)0, c, /*reuse_a=*/false, /*reuse_b=*/false);
    }

    int rbase = m0 + ((lane >> 4) << 3);                // rows m0+v (lanes<16) / m0+8+v
    #pragma unroll
    for (int v = 0; v < 8; ++v) {
        int row = rbase + v;
        out[(size_t)row * D + n0 + col] = c[v] * gate[row];
    }
}

extern "C" void kernel_launch(void* const* d_in, const int* in_sizes, int n_in,
                              void* d_out, int out_size, void* d_ws, size_t ws_size,
                              hipStream_t stream) {
    const float* x   = (const float*)d_in[0];
    const int*   ei  = (const int*)d_in[1];   // [2,E]: src = ei[0..E), dst = ei[E..2E)
    const float* w   = (const float*)d_in[2];
    // d_in[3] node_graph_index unused: graphs are contiguous, equal-sized
    const float* Wf  = (const float*)d_in[4];
    const float* Wsc = (const float*)d_in[5];
    float* out = (float*)d_out;

    // workspace layout (floats unless noted): ~19 MB total
    float* deg   = (float*)d_ws;                  // NUM_NODES
    float* sval  = deg   + NUM_NODES;             // NUM_NODES
    float* score = sval  + NUM_NODES;             // NUM_NODES
    float* gate  = score + NUM_NODES;             // GK
    float* agg   = gate  + GK;                    // GK * D
    int*   rank  = (int*)(agg + (size_t)GK * D);  // NUM_NODES ints

    init_kernel<<<(GK * D + 255) / 256, 256, 0, stream>>>(deg, score, rank, agg);
    deg_kernel<<<NUM_EDGES / 256, 256, 0, stream>>>(ei, w, deg);
    score_pre_kernel<<<NUM_NODES / 8, 256, 0, stream>>>(x, Wsc, sval);
    edge_score_kernel<<<NUM_EDGES / 256, 256, 0, stream>>>(ei, w, deg, sval, score);
    topk_kernel<<<G, N_PER_G, 0, stream>>>(score, rank, gate);
    feat_scatter_kernel<<<NUM_EDGES / 8, 256, 0, stream>>>(ei, w, deg, x, rank, agg);
    gemm_gate_kernel<<<GK / 16, 256, 0, stream>>>(agg, Wf, gate, out);
}